// HierarchicalRenderer_47536698032149
// MI455X (gfx1250) — compile-verified
//
#include <hip/hip_runtime.h>

#define NS 64   // coarse samples per ray
#define NF 128  // fine samples per ray
#define WPB 8   // rays (waves) per 256-thread block

#if __has_builtin(__builtin_amdgcn_mov_dpp8) && __has_builtin(__builtin_amdgcn_update_dpp) && __has_builtin(__builtin_amdgcn_readlane)
#define HAVE_DPP 1
#else
#define HAVE_DPP 0
#endif

// ---------- cross-lane helpers (VALU/DPP path, LDS pipe only for xor16) ----------
template <int M>
__device__ __forceinline__ float lane_xor(float x) {
  if constexpr (M == 16) {
    return __shfl_xor(x, 16, 32);  // ds_swizzle SWAPX16 (only non-DPP stride)
  } else {
#if HAVE_DPP
    int xi = __float_as_int(x);
    int r;
    if constexpr (M <= 4) {
      // DPP8: arbitrary permute within groups of 8 lanes -> exact xor
      r = __builtin_amdgcn_mov_dpp8(
          xi, (0 ^ M) | ((1 ^ M) << 3) | ((2 ^ M) << 6) | ((3 ^ M) << 9) |
                  ((4 ^ M) << 12) | ((5 ^ M) << 15) | ((6 ^ M) << 18) |
                  ((7 ^ M) << 21));
    } else {
      // M == 8: ROW_ROR:8 within row of 16  ==  lane ^ 8
      r = __builtin_amdgcn_update_dpp(xi, xi, 0x128, 0xF, 0xF, false);
    }
    return __int_as_float(r);
#else
    return __shfl_xor(x, M, 32);
#endif
  }
}

__device__ __forceinline__ float rdlane(float x, int l) {
#if HAVE_DPP
  return __int_as_float(__builtin_amdgcn_readlane(__float_as_int(x), l));
#else
  return __shfl(x, l, 32);
#endif
}

#if HAVE_DPP
template <int D>
__device__ __forceinline__ float row_shr_zero(float x) {  // shift-right D in row16, 0-fill
  return __int_as_float(__builtin_amdgcn_update_dpp(0, __float_as_int(x),
                                                    0x110 + D, 0xF, 0xF, true));
}
#endif

__device__ __forceinline__ float wave32_incl_scan(float x, int lane) {
#if HAVE_DPP
  x += row_shr_zero<1>(x);
  x += row_shr_zero<2>(x);
  x += row_shr_zero<4>(x);
  x += row_shr_zero<8>(x);                 // inclusive within each row of 16
  const float t = rdlane(x, 15);           // row0 total (uniform)
  if (lane >= 16) x += t;
  return x;
#else
#pragma unroll
  for (int d = 1; d < 32; d <<= 1) {
    float t = __shfl_up(x, d, 32);
    if (lane >= d) x += t;
  }
  return x;
#endif
}

__device__ __forceinline__ float row16_sum(float x) {
  x += lane_xor<1>(x);
  x += lane_xor<2>(x);
  x += lane_xor<4>(x);
  x += lane_xor<8>(x);
  return x;  // every lane holds its row-of-16 sum
}

// ---------- bitonic network, lane-major layout: element e = lane*NK + k ----------
__device__ __forceinline__ void cmpswap(float& a, float& b, bool desc) {
  float lo = fminf(a, b), hi = fmaxf(a, b);
  a = desc ? hi : lo;
  b = desc ? lo : hi;
}

template <int SIZE, int STRIDE, int NK>
__device__ __forceinline__ void bpass(float* v, int lane) {
  if constexpr (STRIDE < NK) {  // register-register compare-exchange
#pragma unroll
    for (int k = 0; k < NK; ++k)
      if (!(k & STRIDE)) {
        bool desc = ((lane * NK + k) & SIZE) != 0;
        cmpswap(v[k], v[k + STRIDE], desc);
      }
  } else {  // cross-lane compare-exchange
    constexpr int LX = STRIDE / NK;
#pragma unroll
    for (int k = 0; k < NK; ++k) {
      bool desc = ((lane * NK + k) & SIZE) != 0;
      float other = lane_xor<LX>(v[k]);
      bool upper = (lane & LX) != 0;
      v[k] = (upper == desc) ? fminf(v[k], other) : fmaxf(v[k], other);
    }
  }
}

template <int SIZE, int STRIDE, int NK>
__device__ __forceinline__ void bmerge(float* v, int lane) {
  bpass<SIZE, STRIDE, NK>(v, lane);
  if constexpr (STRIDE > 1) bmerge<SIZE, (STRIDE >> 1), NK>(v, lane);
}

template <int SIZE, int NTOT, int NK>
__device__ __forceinline__ void bsort(float* v, int lane) {
  bmerge<SIZE, (SIZE >> 1), NK>(v, lane);
  if constexpr (SIZE < NTOT) bsort<(SIZE << 1), NTOT, NK>(v, lane);
}

// ---------- kernel: one ray per wave32 ----------
__global__ __launch_bounds__(256) void hier_render_kernel(
    const float* __restrict__ rgb,     // [N,64,3]
    const float* __restrict__ sigma,   // [N,64]
    const float* __restrict__ zvg,     // [N,64] sorted
    const float* __restrict__ rays_d,  // [N,3]
    const float* __restrict__ uu,      // [N,128]
    float* __restrict__ o_rgb, float* __restrict__ o_depth,
    float* __restrict__ o_acc, float* __restrict__ o_disp,
    float* __restrict__ o_w, float* __restrict__ o_zc, int N) {
  __shared__ __align__(16) float sh_z[WPB][NS];
  __shared__ float sh_cdf[WPB][NS + 2];
  __shared__ __align__(16) float sh_zf[WPB][NF];

  const int lane = threadIdx.x & 31;
  const int wid = threadIdx.x >> 5;
  const int ray = blockIdx.x * WPB + wid;
  if (ray >= N) return;

  // ---- CDNA5 async DMA: stage z row straight into LDS (8B/lane = 256B/wave)
  {
    unsigned loff = (unsigned)(size_t)(void*)(&sh_z[wid][2 * lane]);
    unsigned long long ga =
        (unsigned long long)(zvg + (size_t)ray * NS + 2 * lane);
    asm volatile("global_load_async_to_lds_b64 %0, %1, off"
                 :
                 : "v"(loff), "v"(ga)
                 : "memory");
  }
  __builtin_prefetch(rgb + (size_t)ray * (3 * NS) + 6 * lane, 0, 0);
  __builtin_prefetch(uu + (size_t)ray * NF + 4 * lane, 0, 0);

  const float2 sg = *(const float2*)(sigma + (size_t)ray * NS + 2 * lane);
  const float dx = rays_d[ray * 3 + 0];
  const float dy = rays_d[ray * 3 + 1];
  const float dz = rays_d[ray * 3 + 2];
  const float nrm = sqrtf(dx * dx + dy * dy + dz * dz);

  asm volatile("s_wait_asynccnt 0" ::: "memory");
  __builtin_amdgcn_wave_barrier();

  const float* zp = sh_z[wid];
  const float z0 = zp[2 * lane];
  const float z1 = zp[2 * lane + 1];
  const float zn = (lane < 31) ? zp[2 * lane + 2] : 0.0f;
  const float d0 = (z1 - z0) * nrm;
  const float d1 = (lane == 31) ? (1e10f * nrm) : ((zn - z1) * nrm);
  const float s0 = fmaxf(sg.x, 0.0f) * d0;
  const float s1 = fmaxf(sg.y, 0.0f) * d1;

  // transmittance via exp(-exclusive cumsum of optical depth)
  const float p = s0 + s1;
  const float P = wave32_incl_scan(p, lane);
  const float c0 = P - p;
  const float c1 = c0 + s0;
  const float T0 = __expf(-c0);
  const float T1 = __expf(-c1);
  const float T2 = __expf(-(c1 + s1));
  const float w0 = T0 - T1;
  const float w1 = T1 - T2;

  {
    float2 wv;
    wv.x = w0;
    wv.y = w1;
    *(float2*)(o_w + (size_t)ray * NS + 2 * lane) = wv;
  }

  // per-ray reductions: rgb(3), depth, acc  (row16 DPP tree + readlane combine)
  const float* rg = rgb + (size_t)ray * (3 * NS) + 6 * lane;
  float ar = w0 * rg[0] + w1 * rg[3];
  float ag = w0 * rg[1] + w1 * rg[4];
  float ab = w0 * rg[2] + w1 * rg[5];
  float ad = w0 * z0 + w1 * z1;
  float aa = w0 + w1;
  ar = row16_sum(ar);
  ag = row16_sum(ag);
  ab = row16_sum(ab);
  ad = row16_sum(ad);
  aa = row16_sum(aa);
  const float arT = ar + rdlane(ar, 16);
  const float agT = ag + rdlane(ag, 16);
  const float abT = ab + rdlane(ab, 16);
  const float adT = ad + rdlane(ad, 16);
  const float aaT = aa + rdlane(aa, 16);
  if (lane == 0) {
    o_rgb[ray * 3 + 0] = arT;
    o_rgb[ray * 3 + 1] = agT;
    o_rgb[ray * 3 + 2] = abT;
    o_depth[ray] = adT;
    o_acc[ray] = aaT;
    o_disp[ray] = 1.0f / fmaxf(1e-10f, adT / aaT);
  }

  // CDF over (weights + 1e-5): 65 entries in LDS
  const float q0 = w0 + 1e-5f;
  const float q1 = w1 + 1e-5f;
  const float Q = wave32_incl_scan(q0 + q1, lane);
  const float total = rdlane(Q, 31);
  const float inv = 1.0f / total;
  float* cdfw = sh_cdf[wid];
  if (lane == 0) cdfw[0] = 0.0f;
  cdfw[2 * lane + 1] = (Q - q1) * inv;
  cdfw[2 * lane + 2] = Q * inv;
  __builtin_amdgcn_wave_barrier();

  // importance sampling: 4 fine samples per lane -> kept in registers for the sort
  const float4 u4 = *(const float4*)(uu + (size_t)ray * NF + 4 * lane);
  const float uf[4] = {u4.x, u4.y, u4.z, u4.w};
  float v[4];
#pragma unroll
  for (int t = 0; t < 4; ++t) {
    const float uv = uf[t];
    int lo = 0, hi = NS + 1;  // searchsorted(right) over cdf[0..64]
#pragma unroll
    for (int it = 0; it < 7; ++it) {
      int mid = (lo + hi) >> 1;
      bool go = lo < hi;
      bool ge = go && (uv >= cdfw[mid]);
      lo = ge ? (mid + 1) : lo;
      hi = (go && !ge) ? mid : hi;
    }
    int below = lo - 1;
    int above = (lo > NS) ? NS : lo;
    const float cb = cdfw[below];
    const float ca = cdfw[above];
    const float zb = zp[(below < NS) ? below : (NS - 1)];
    const float za = zp[(above < NS) ? above : (NS - 1)];
    float den = ca - cb;
    if (den < 1e-5f) den = 1.0f;
    const float tt = (uv - cb) / den;
    v[t] = zb + tt * (za - zb);
  }

  // sort 128 fine depths ascending (lane-major, sort input order is irrelevant)
  bsort<2, NF, 4>(v, lane);

  // stash sorted zf, then build 256-long bitonic sequence (lane-major e = lane*8+k):
  //   e in [0,64)   : z ascending       (lanes 0..7)
  //   e in [64,128) : +big pad          (lanes 8..15)  -- head of descending half
  //   e in [128,256): zf descending     (lanes 16..31)
  float* zfw = sh_zf[wid];
  *(float4*)(&zfw[lane * 4]) = make_float4(v[0], v[1], v[2], v[3]);
  __builtin_amdgcn_wave_barrier();

  float w[8];
  if (lane < 8) {
    const float4 A = *(const float4*)(&zp[lane * 8]);
    const float4 B = *(const float4*)(&zp[lane * 8 + 4]);
    w[0] = A.x; w[1] = A.y; w[2] = A.z; w[3] = A.w;
    w[4] = B.x; w[5] = B.y; w[6] = B.z; w[7] = B.w;
  } else if (lane < 16) {
#pragma unroll
    for (int k = 0; k < 8; ++k) w[k] = 1e30f;
  } else {
    const int b0 = 120 - (lane - 16) * 8;  // reversed contiguous block
    const float4 A = *(const float4*)(&zfw[b0]);
    const float4 B = *(const float4*)(&zfw[b0 + 4]);
    w[0] = B.w; w[1] = B.z; w[2] = B.y; w[3] = B.x;
    w[4] = A.w; w[5] = A.z; w[6] = A.y; w[7] = A.x;
  }

  // single ascending bitonic merge of the 256-long bitonic sequence
  bmerge<256, 128, 8>(w, lane);

  // first 192 elements = sorted concat (pads live in lanes 24..31)
  if (lane < 24) {
    float* ozc = o_zc + (size_t)ray * (NS + NF) + lane * 8;
    *(float4*)(ozc) = make_float4(w[0], w[1], w[2], w[3]);
    *(float4*)(ozc + 4) = make_float4(w[4], w[5], w[6], w[7]);
  }
}

extern "C" void kernel_launch(void* const* d_in, const int* in_sizes, int n_in,
                              void* d_out, int out_size, void* d_ws,
                              size_t ws_size, hipStream_t stream) {
  const float* rgb = (const float*)d_in[0];
  const float* sigma = (const float*)d_in[1];
  const float* zv = (const float*)d_in[2];
  const float* rd = (const float*)d_in[3];
  const float* u = (const float*)d_in[4];
  const int N = in_sizes[1] / NS;  // sigma is [N,64]

  float* out = (float*)d_out;
  float* o_rgb = out;                    // [N,3]
  float* o_depth = out + (size_t)3 * N;  // [N]
  float* o_acc = out + (size_t)4 * N;    // [N]
  float* o_disp = out + (size_t)5 * N;   // [N]
  float* o_w = out + (size_t)6 * N;      // [N,64]
  float* o_zc = out + (size_t)70 * N;    // [N,192]

  dim3 block(256);
  dim3 grid((N + WPB - 1) / WPB);
  hipLaunchKernelGGL(hier_render_kernel, grid, block, 0, stream, rgb, sigma, zv,
                     rd, u, o_rgb, o_depth, o_acc, o_disp, o_w, o_zc, N);
}